// ELLGAT_51797305589896
// MI455X (gfx1250) — compile-verified
//
#include <hip/hip_runtime.h>

typedef __attribute__((ext_vector_type(2))) float v2f;
typedef __attribute__((ext_vector_type(4))) float f4;
typedef __attribute__((ext_vector_type(8))) float v8f;

#define N_NODES  50000
#define MAX_NBR  32
#define Q_FEAT   128
#define OUT_FEAT 16
#define HEADS    4

// ---------------------------------------------------------------------------
// Phase 1: Wp = W[h] (16x128) x X (128xN)  ->  stored node-major [H][N][16]
// One wave (32 lanes) per 16-node tile per head. fp32 WMMA 16x16x4, 32 steps.
// ---------------------------------------------------------------------------
__global__ __launch_bounds__(32) void ellgat_proj_wmma(
    const float* __restrict__ W,    // [HEADS][16][128]
    const float* __restrict__ X,    // [128][N]
    float* __restrict__ out)        // [HEADS][N][16]  (feature-contiguous)
{
    const int tile = blockIdx.x;            // 0..3124 (N/16 exact)
    const int h    = blockIdx.y;            // head
    const int lane = threadIdx.x;           // 0..31
    const int m    = lane & 15;             // row (=out feature) / col (=node)
    const int khi  = (lane >> 4) & 1;       // upper-half lanes hold K+2,K+3
    const int n    = tile * 16 + m;         // node index for B/C fragments

    const float* __restrict__ Wh = W + h * (OUT_FEAT * Q_FEAT);

    v8f c = {0.f, 0.f, 0.f, 0.f, 0.f, 0.f, 0.f, 0.f};

#pragma unroll
    for (int k0 = 0; k0 < Q_FEAT; k0 += 4) {
        const int k = k0 + 2 * khi;
        // A fragment: 16x4 fp32. VGPR0: K=k, VGPR1: K=k+1 (per lane half).
        v2f a;
        a.x = Wh[m * Q_FEAT + k];
        a.y = Wh[m * Q_FEAT + k + 1];
        // B fragment: 4x16 fp32, rows striped across lanes (mirrors A).
        v2f b;
        b.x = X[(size_t)k       * N_NODES + n];
        b.y = X[(size_t)(k + 1) * N_NODES + n];
        c = __builtin_amdgcn_wmma_f32_16x16x4_f32(
                /*neg_a=*/false, a, /*neg_b=*/false, b,
                /*c_mod=*/(short)0, c, /*reuse_a=*/false, /*reuse_b=*/false);
    }

    // C/D layout: VGPR j -> M = j + 8*khi, N = lane&15. Store transposed to
    // node-major [h][n][o]: each lane writes a contiguous 32B span (o dim).
    const int mBase = khi * 8;
    float* __restrict__ dst = out + ((size_t)h * N_NODES + n) * OUT_FEAT + mBase;
    f4 lo = {c[0], c[1], c[2], c[3]};
    f4 hi = {c[4], c[5], c[6], c[7]};
    *(f4*)(dst)     = lo;
    *(f4*)(dst + 4) = hi;
}

// ---------------------------------------------------------------------------
// Phase 2: gather + masked softmax + weighted combine.
// wave32: lane = neighbor slot k (MAX_NBR==32). One wave per (head, node).
// ---------------------------------------------------------------------------
__global__ __launch_bounds__(256) void ellgat_attn(
    const int*   __restrict__ adj,   // [N][32]
    const float* __restrict__ Qp,    // [H][N][16]
    const float* __restrict__ Kp,    // [H][N][16]
    const float* __restrict__ aw,    // [H][16]
    float*       __restrict__ out)   // [H][16][N]
{
    const int lane = threadIdx.x;                       // neighbor slot
    const int node = blockIdx.x * blockDim.y + threadIdx.y;
    const int h    = blockIdx.y;
    if (node >= N_NODES) return;

    const int  j     = adj[(size_t)node * MAX_NBR + lane];
    const bool valid = (j >= 0);
    const int  jj    = valid ? j : 0;

    // Gather neighbor key features: 64B contiguous per lane (L2 resident).
    const f4* __restrict__ kgp = (const f4*)(Kp + ((size_t)h * N_NODES + jj) * OUT_FEAT);
    f4 kg0 = kgp[0], kg1 = kgp[1], kg2 = kgp[2], kg3 = kgp[3];
    // Query features for this node (uniform across the wave).
    const f4* __restrict__ qpp = (const f4*)(Qp + ((size_t)h * N_NODES + node) * OUT_FEAT);
    f4 q0 = qpp[0], q1 = qpp[1], q2 = qpp[2], q3 = qpp[3];

    f4 p0 = q0 * kg0, p1 = q1 * kg1, p2 = q2 * kg2, p3 = q3 * kg3;

    float x[OUT_FEAT];
    x[0]=p0.x; x[1]=p0.y; x[2]=p0.z; x[3]=p0.w;
    x[4]=p1.x; x[5]=p1.y; x[6]=p1.z; x[7]=p1.w;
    x[8]=p2.x; x[9]=p2.y; x[10]=p2.z; x[11]=p2.w;
    x[12]=p3.x; x[13]=p3.y; x[14]=p3.z; x[15]=p3.w;
#pragma unroll
    for (int o = 0; o < OUT_FEAT; ++o)            // leaky_relu, slope 0.01
        x[o] = (x[o] > 0.f) ? x[o] : 0.01f * x[o];

    const float NEG = -3.0e38f;
    float alpha = 0.f;                            // sum_w attn[w,lane]*aw[w]
#pragma unroll
    for (int o = 0; o < OUT_FEAT; ++o) {
        float xm = valid ? x[o] : NEG;
        float mx = xm;                            // wave max over neighbors
#pragma unroll
        for (int s = 16; s > 0; s >>= 1)
            mx = fmaxf(mx, __shfl_xor(mx, s, 32));
        float e = valid ? __expf(x[o] - mx) : 0.f;
        float ssum = e;                           // wave sum
#pragma unroll
        for (int s = 16; s > 0; s >>= 1)
            ssum += __shfl_xor(ssum, s, 32);
        alpha += (e / ssum) * aw[h * OUT_FEAT + o];
    }

    // out[h,o,node] = sum over neighbors of Xm[o,k] * alpha[k]
    float res = 0.f;
#pragma unroll
    for (int o = 0; o < OUT_FEAT; ++o) {
        float r = (valid ? x[o] : 0.f) * alpha;
#pragma unroll
        for (int s = 16; s > 0; s >>= 1)
            r += __shfl_xor(r, s, 32);            // all lanes hold sum
        if (lane == o) res = r;                   // lane o keeps feature o
    }
    if (lane < OUT_FEAT)
        out[((size_t)h * OUT_FEAT + lane) * N_NODES + node] = res;
}

// ---------------------------------------------------------------------------
extern "C" void kernel_launch(void* const* d_in, const int* in_sizes, int n_in,
                              void* d_out, int out_size, void* d_ws, size_t ws_size,
                              hipStream_t stream) {
    const int*   adj = (const int*)  d_in[0];
    const float* Q   = (const float*)d_in[1];
    const float* K   = (const float*)d_in[2];
    const float* qw  = (const float*)d_in[3];
    const float* kw  = (const float*)d_in[4];
    const float* aw  = (const float*)d_in[5];
    float* out = (float*)d_out;

    float* Qp = (float*)d_ws;                                  // [H][N][16]
    float* Kp = Qp + (size_t)HEADS * N_NODES * OUT_FEAT;       // [H][N][16]

    dim3 g1(N_NODES / 16, HEADS);                              // 3125 x 4
    ellgat_proj_wmma<<<g1, dim3(32, 1, 1), 0, stream>>>(qw, Q, Qp);
    ellgat_proj_wmma<<<g1, dim3(32, 1, 1), 0, stream>>>(kw, K, Kp);

    dim3 g2((N_NODES + 7) / 8, HEADS);                         // 8 waves/block
    ellgat_attn<<<g2, dim3(32, 8, 1), 0, stream>>>(adj, Qp, Kp, aw, out);
}